// RBFKernel_65481071407813
// MI455X (gfx1250) — compile-verified
//
#include <hip/hip_runtime.h>

typedef __attribute__((ext_vector_type(2))) float v2f;
typedef __attribute__((ext_vector_type(4))) float v4f;
typedef __attribute__((ext_vector_type(8))) float v8f;

#define NROWS 8192
#define DIM   64
#define GAMMA 1.0f

// ---------------------------------------------------------------------------
// Prologue: per-row squared norms of x and y into workspace.
// 8192 threads, one row each; rows stream through L0 (2 cachelines/row).
// ---------------------------------------------------------------------------
__global__ __launch_bounds__(256) void rbf_rowsq(const float* __restrict__ x,
                                                 const float* __restrict__ y,
                                                 float* __restrict__ xsq,
                                                 float* __restrict__ ysq) {
    const int row = blockIdx.x * 256 + threadIdx.x;
    const float4* xr = (const float4*)(x + (size_t)row * DIM);
    const float4* yr = (const float4*)(y + (size_t)row * DIM);
    float sx = 0.0f, sy = 0.0f;
#pragma unroll
    for (int i = 0; i < DIM / 4; ++i) {
        float4 a = xr[i];
        float4 b = yr[i];
        sx += a.x * a.x + a.y * a.y + a.z * a.z + a.w * a.w;
        sy += b.x * b.x + b.y * b.y + b.z * b.z + b.w * b.w;
    }
    xsq[row] = sx;
    ysq[row] = sy;
}

// ---------------------------------------------------------------------------
// Main: each wave computes a 64x64 output tile with 4x4 WMMA f32 16x16x4
// accumulators (register blocked; ~8 B of L2 input traffic per output).
//
// A-fragment layout (16x4 f32): lane = M (mod 16); VGPR0/1 hold K = 2*half
// and K = 2*half+1 where half = lane>>4. B mirrors A with lane = N since
// B[k][n] = y[n][k] for the x·yT GEMM.
// C/D layout: VGPR e -> M = e + 8*half, N = lane&15.
// ---------------------------------------------------------------------------
__global__ __launch_bounds__(256) void rbf_wmma(const float* __restrict__ x,
                                                const float* __restrict__ y,
                                                const float* __restrict__ xsq,
                                                const float* __restrict__ ysq,
                                                float* __restrict__ out) {
    const int lane = threadIdx.x & 31;
    const int wave = threadIdx.x >> 5;
    const int half = lane >> 4;
    const int lrow = lane & 15;

    const int n0 = blockIdx.x * 64;
    const int m0 = (blockIdx.y * 8 + wave) * 64;

    // Per-lane base pointers for A (rows of x) and B (rows of y).
    const float* ax = x + (size_t)(m0 + lrow) * DIM + 2 * half;
    const float* by = y + (size_t)(n0 + lrow) * DIM + 2 * half;

    v8f acc[4][4] = {};

#pragma unroll 4
    for (int k = 0; k < DIM / 4; ++k) {
        v2f a[4], b[4];
#pragma unroll
        for (int i = 0; i < 4; ++i)
            a[i] = *(const v2f*)(ax + (size_t)(16 * i) * DIM + 4 * k);
#pragma unroll
        for (int j = 0; j < 4; ++j)
            b[j] = *(const v2f*)(by + (size_t)(16 * j) * DIM + 4 * k);

#pragma unroll
        for (int i = 0; i < 4; ++i)
#pragma unroll
            for (int j = 0; j < 4; ++j)
                acc[i][j] = __builtin_amdgcn_wmma_f32_16x16x4_f32(
                    /*neg_a=*/false, a[i], /*neg_b=*/false, b[j],
                    /*c_mod=*/(short)0, acc[i][j],
                    /*reuse_a=*/false, /*reuse_b=*/false);
    }

    // Epilogue: dist^2 = |x|^2 + |y|^2 - 2 x.y, floor at 0, exp(-gamma*d).
    const float neg_g_log2e = -1.4426950408889634f * GAMMA;

    float ysq_l[4];
#pragma unroll
    for (int j = 0; j < 4; ++j) ysq_l[j] = ysq[n0 + 16 * j + lrow];

#pragma unroll
    for (int i = 0; i < 4; ++i) {
        const int mbase = m0 + 16 * i + 8 * half;
        v4f xs_lo = *(const v4f*)(xsq + mbase);
        v4f xs_hi = *(const v4f*)(xsq + mbase + 4);
#pragma unroll
        for (int j = 0; j < 4; ++j) {
            float* orow = out + (size_t)mbase * NROWS + n0 + 16 * j + lrow;
#pragma unroll
            for (int e = 0; e < 8; ++e) {
                const float xs = (e < 4) ? xs_lo[e] : xs_hi[e - 4];
                float d = xs + ysq_l[j] - 2.0f * acc[i][j][e];
                d = fmaxf(d, 0.0f);
                // exp(-g*d) = exp2(-g*log2(e)*d) -> v_exp_f32
                const float r = __builtin_amdgcn_exp2f(neg_g_log2e * d);
                // Output is write-once: non-temporal keeps L2 for inputs.
                __builtin_nontemporal_store(r, orow + (size_t)e * NROWS);
            }
        }
    }
}

// ---------------------------------------------------------------------------
extern "C" void kernel_launch(void* const* d_in, const int* in_sizes, int n_in,
                              void* d_out, int out_size, void* d_ws, size_t ws_size,
                              hipStream_t stream) {
    const float* x = (const float*)d_in[0];
    const float* y = (const float*)d_in[1];
    float* out = (float*)d_out;

    float* xsq = (float*)d_ws;        // 8192 floats
    float* ysq = xsq + NROWS;         // 8192 floats (64 KB total workspace)

    rbf_rowsq<<<NROWS / 256, 256, 0, stream>>>(x, y, xsq, ysq);

    // 128 n-tiles of 64; 128 m-tiles of 64 spread as 16 blocks x 8 waves.
    dim3 grid(NROWS / 64, NROWS / (64 * 8));
    rbf_wmma<<<grid, 256, 0, stream>>>(x, y, xsq, ysq, out);
}